// PG_88527865905287
// MI455X (gfx1250) — compile-verified
//
#include <hip/hip_runtime.h>
#include <math.h>

// ---------------- problem constants ----------------
#define BB   8
#define NN   16384
#define DL   128
#define DD   640      // D
#define DH   160      // D/4
#define KK   8192     // top-k
#define DIN2 771      // 3 + 128 + 640

typedef __attribute__((ext_vector_type(2))) float v2f;
typedef __attribute__((ext_vector_type(8))) float v8f;

__device__ __forceinline__ v8f wmma4(v2f a, v2f b, v8f c) {
  // D = A(16x4,f32) * B(4x16,f32) + C(16x16,f32)
  return __builtin_amdgcn_wmma_f32_16x16x4_f32(
      /*neg_a=*/false, a, /*neg_b=*/false, b,
      /*c_mod=*/(short)0, c, /*reuse_a=*/false, /*reuse_b=*/false);
}

__device__ __forceinline__ float lrelu(float v) { return v >= 0.f ? v : 0.2f * v; }

// order-preserving float<->uint encoding (for max via unsigned atomicMax)
__device__ __forceinline__ unsigned fenc(float f) {
  unsigned u = __float_as_uint(f);
  return (u & 0x80000000u) ? ~u : (u | 0x80000000u);
}
__device__ __forceinline__ float fdec(unsigned e) {
  unsigned u = (e & 0x80000000u) ? (e & 0x7FFFFFFFu) : ~e;
  return __uint_as_float(u);
}

// =====================================================================
// Kernel 1: score[b][n] = sigmoid(w2 . relu(W1 x + b1))
// one wave per 16-point tile, WMMA f32 16x16x4
// =====================================================================
__global__ __launch_bounds__(256) void score_kernel(
    const float* __restrict__ x, const float* __restrict__ w1,
    const float* __restrict__ b1, const float* __restrict__ w2,
    float* __restrict__ score) {
  const int lane = threadIdx.x & 31;
  const int wv   = threadIdx.x >> 5;
  const int lm   = lane & 15;   // A-row / B-col within tile
  const int kh   = lane >> 4;   // half-wave: K sub-offset selector
  const int tile = blockIdx.x * 8 + wv;
  const int p0   = tile * 16;
  const int b    = p0 / NN;
  const int n    = (p0 % NN) + lm;
  const float* xb = x + (size_t)b * DD * NN;

  float part = 0.f;
  for (int mt = 0; mt < 10; ++mt) {
    v8f acc = {0.f, 0.f, 0.f, 0.f, 0.f, 0.f, 0.f, 0.f};
    const float* Wrow = w1 + (size_t)(mt * 16 + lm) * DD;
    for (int k0 = 0; k0 < DD; k0 += 4) {
      const int ka = k0 + 2 * kh;
      v2f a, bv;
      a.x = Wrow[ka];
      a.y = Wrow[ka + 1];
      const float* xc = xb + (size_t)ka * NN + n;
      bv.x = xc[0];
      bv.y = xc[NN];
      acc = wmma4(a, bv, acc);
    }
#pragma unroll
    for (int r = 0; r < 8; ++r) {
      const int m = mt * 16 + kh * 8 + r;   // D-layout row for vgpr r
      float hv = acc[r] + b1[m];
      hv = hv > 0.f ? hv : 0.f;
      part += w2[m] * hv;
    }
  }
  const float tot = part + __shfl_xor(part, 16, 32);
  if (kh == 0) {
    score[(size_t)b * NN + n] = 1.f / (1.f + __expf(-tot));
  }
}

// =====================================================================
// Kernel 2: exact top-k with ordering — full bitonic sort of one batch
// row (16384 64-bit keys = 128 KB) entirely in WGP LDS.
// key = score_bits<<32 | (0xFFFFFFFF - idx)  → descending sort gives
// score-descending order with lower-index-first tiebreak (matches top_k).
// =====================================================================
__global__ __launch_bounds__(1024) void topk_sort_kernel(
    const float* __restrict__ score, int* __restrict__ idxk,
    float* __restrict__ sck) {
  __shared__ unsigned long long keys[NN];   // 128 KB
  const int b = blockIdx.x;
  for (int i = threadIdx.x; i < NN; i += blockDim.x) {
    const unsigned sb = __float_as_uint(score[(size_t)b * NN + i]);
    keys[i] = ((unsigned long long)sb << 32) | (unsigned)(0xFFFFFFFFu - (unsigned)i);
  }
  __syncthreads();
  for (int k = 2; k <= NN; k <<= 1) {
    for (int j = k >> 1; j > 0; j >>= 1) {
      for (int i = threadIdx.x; i < NN; i += blockDim.x) {
        const int ixj = i ^ j;
        if (ixj > i) {
          const unsigned long long a = keys[i];
          const unsigned long long c = keys[ixj];
          const bool up = ((i & k) == 0);   // descending overall
          if (up ? (a < c) : (a > c)) { keys[i] = c; keys[ixj] = a; }
        }
      }
      __syncthreads();
    }
  }
  for (int t = threadIdx.x; t < KK; t += blockDim.x) {
    const unsigned long long kk = keys[t];
    idxk[(size_t)b * KK + t] = (int)(0xFFFFFFFFu - (unsigned)(kk & 0xFFFFFFFFu));
    sck[(size_t)b * KK + t]  = __uint_as_float((unsigned)(kk >> 32));
  }
}

// =====================================================================
// Kernel 3: gather pos/f_local/feat + d1 MLP + max-pool (atomic)
// one thread per output point (b,t)
// =====================================================================
__global__ __launch_bounds__(256) void gather_d1_kernel(
    const float* __restrict__ pos, const float* __restrict__ flocal,
    const float* __restrict__ x, const int* __restrict__ idxk,
    const float* __restrict__ sck,
    const float* __restrict__ scw, const float* __restrict__ scb,
    const float* __restrict__ c1w, const float* __restrict__ c1b,
    const float* __restrict__ c2aw, const float* __restrict__ c2ab,
    const float* __restrict__ c2bw, const float* __restrict__ c2bb,
    float* __restrict__ posk, float* __restrict__ out_flocal,
    float* __restrict__ out_feat, unsigned* __restrict__ gbits) {
  __shared__ unsigned smax[128];
  const int gid = blockIdx.x * 256 + threadIdx.x;   // 0..65535, one b per block
  const int b = gid / KK;
  const int t = gid % KK;
  if (threadIdx.x < 128) smax[threadIdx.x] = 0u;
  __syncthreads();

  const int   idx = idxk[(size_t)b * KK + t];
  const float s   = sck[(size_t)b * KK + t];

  float p[3];
#pragma unroll
  for (int r = 0; r < 3; ++r) {
    p[r] = pos[((size_t)b * 3 + r) * NN + idx];
    posk[((size_t)b * 3 + r) * KK + t] = p[r];
  }
  for (int c = 0; c < DL; ++c)
    out_flocal[((size_t)b * DL + c) * KK + t] = flocal[((size_t)b * DL + c) * NN + idx];
  for (int c = 0; c < DD; ++c)
    out_feat[((size_t)b * DD + c) * KK + t] = x[((size_t)b * DD + c) * NN + idx] * s;

  // d1 residual MLP on p[3]
  float h1[64];
  for (int o = 0; o < 64; ++o)
    h1[o] = lrelu(c1b[o] + c1w[o * 3] * p[0] + c1w[o * 3 + 1] * p[1] + c1w[o * 3 + 2] * p[2]);
  float h2[128];
  for (int o = 0; o < 128; ++o) {
    float v = c2ab[o];
    for (int i = 0; i < 64; ++i) v += c2aw[o * 64 + i] * h1[i];
    h2[o] = lrelu(v);
  }
  for (int o = 0; o < 128; ++o) {
    float v = c2bb[o] + scb[o] +
              scw[o * 3] * p[0] + scw[o * 3 + 1] * p[1] + scw[o * 3 + 2] * p[2];
    for (int i = 0; i < 128; ++i) v += c2bw[o * 128 + i] * h2[i];
    atomicMax(&smax[o], fenc(v));    // LDS ds_max_u32
  }
  __syncthreads();
  if (threadIdx.x < 128) atomicMax(&gbits[b * 128 + threadIdx.x], smax[threadIdx.x]);
}

__global__ void init_g_kernel(unsigned* __restrict__ gbits) {
  const int i = blockIdx.x * blockDim.x + threadIdx.x;
  if (i < BB * 128) gbits[i] = 0u;   // below fenc(-inf)
}
__global__ void fin_g_kernel(const unsigned* __restrict__ gbits, float* __restrict__ g) {
  const int i = blockIdx.x * blockDim.x + threadIdx.x;
  if (i < BB * 128) g[i] = fdec(gbits[i]);
}

// =====================================================================
// Kernel 4: d2 residual MLP on cat = [pos_k(3), g(128), feat_k(640)]
// + c3 head + residual. One wave per 16-column tile; WMMA throughout,
// intermediates staged in per-wave LDS (in-order DS per wave).
// =====================================================================
__device__ __forceinline__ float catf(const float* __restrict__ posk,
                                      const float* __restrict__ g,
                                      const float* __restrict__ feat,
                                      int b, int c, int t) {
  if (c >= DIN2) return 0.f;                                   // K-pad
  if (c < 3)     return posk[((size_t)b * 3 + c) * KK + t];
  if (c < 131)   return g[b * 128 + (c - 3)];
  return feat[((size_t)b * DD + (c - 131)) * KK + t];
}

__global__ __launch_bounds__(256) void d2_kernel(
    const float* __restrict__ posk, const float* __restrict__ g,
    const float* __restrict__ feat,
    const float* __restrict__ scw, const float* __restrict__ scb,
    const float* __restrict__ c1w, const float* __restrict__ c1b,
    const float* __restrict__ c2aw, const float* __restrict__ c2ab,
    const float* __restrict__ c2bw, const float* __restrict__ c2bb,
    const float* __restrict__ c3w, const float* __restrict__ c3b,
    float* __restrict__ pos_out) {
  __shared__ float stage[8][128][16];   // 64 KB, per-wave slabs
  const int lane = threadIdx.x & 31;
  const int wv   = threadIdx.x >> 5;
  const int lm   = lane & 15;
  const int kh   = lane >> 4;
  const int tile = blockIdx.x * 8 + wv;
  const int q0   = tile * 16;
  const int b    = q0 / KK;
  const int t    = (q0 % KK) + lm;

  // --- fused c1 (128 rows) + sc (64 rows) over K=771 (padded 772) ---
  v8f c1acc[8], scacc[4];
#pragma unroll
  for (int i = 0; i < 8; ++i) c1acc[i] = (v8f){0.f,0.f,0.f,0.f,0.f,0.f,0.f,0.f};
#pragma unroll
  for (int i = 0; i < 4; ++i) scacc[i] = (v8f){0.f,0.f,0.f,0.f,0.f,0.f,0.f,0.f};

  for (int k0 = 0; k0 < 772; k0 += 4) {
    const int ka = k0 + 2 * kh;
    v2f bv;
    bv.x = catf(posk, g, feat, b, ka, t);
    bv.y = catf(posk, g, feat, b, ka + 1, t);
#pragma unroll
    for (int mt = 0; mt < 8; ++mt) {
      const int row = mt * 16 + lm;
      v2f a;
      a.x = (ka     < DIN2) ? c1w[(size_t)row * DIN2 + ka]     : 0.f;
      a.y = (ka + 1 < DIN2) ? c1w[(size_t)row * DIN2 + ka + 1] : 0.f;
      c1acc[mt] = wmma4(a, bv, c1acc[mt]);
    }
#pragma unroll
    for (int mt = 0; mt < 4; ++mt) {
      const int row = mt * 16 + lm;
      v2f a;
      a.x = (ka     < DIN2) ? scw[(size_t)row * DIN2 + ka]     : 0.f;
      a.y = (ka + 1 < DIN2) ? scw[(size_t)row * DIN2 + ka + 1] : 0.f;
      scacc[mt] = wmma4(a, bv, scacc[mt]);
    }
  }

  // h1 = lrelu(c1 + b) -> stage rows 0..127
#pragma unroll
  for (int mt = 0; mt < 8; ++mt)
#pragma unroll
    for (int r = 0; r < 8; ++r) {
      const int m = mt * 16 + kh * 8 + r;
      stage[wv][m][lm] = lrelu(c1acc[mt][r] + c1b[m]);
    }

  // --- c2a: 64 x 128 ---
  v8f aacc[4];
#pragma unroll
  for (int i = 0; i < 4; ++i) aacc[i] = (v8f){0.f,0.f,0.f,0.f,0.f,0.f,0.f,0.f};
  for (int k0 = 0; k0 < 128; k0 += 4) {
    const int ka = k0 + 2 * kh;
    v2f bv;
    bv.x = stage[wv][ka][lm];
    bv.y = stage[wv][ka + 1][lm];
#pragma unroll
    for (int mt = 0; mt < 4; ++mt) {
      const int row = mt * 16 + lm;
      v2f a;
      a.x = c2aw[(size_t)row * 128 + ka];
      a.y = c2aw[(size_t)row * 128 + ka + 1];
      aacc[mt] = wmma4(a, bv, aacc[mt]);
    }
  }
  // h2 = lrelu(c2a + b) -> stage rows 0..63 (per-wave DS is in-order)
#pragma unroll
  for (int mt = 0; mt < 4; ++mt)
#pragma unroll
    for (int r = 0; r < 8; ++r) {
      const int m = mt * 16 + kh * 8 + r;
      stage[wv][m][lm] = lrelu(aacc[mt][r] + c2ab[m]);
    }

  // --- c2b: 64 x 64 ---
  v8f bacc[4];
#pragma unroll
  for (int i = 0; i < 4; ++i) bacc[i] = (v8f){0.f,0.f,0.f,0.f,0.f,0.f,0.f,0.f};
  for (int k0 = 0; k0 < 64; k0 += 4) {
    const int ka = k0 + 2 * kh;
    v2f bv;
    bv.x = stage[wv][ka][lm];
    bv.y = stage[wv][ka + 1][lm];
#pragma unroll
    for (int mt = 0; mt < 4; ++mt) {
      const int row = mt * 16 + lm;
      v2f a;
      a.x = c2bw[(size_t)row * 64 + ka];
      a.y = c2bw[(size_t)row * 64 + ka + 1];
      bacc[mt] = wmma4(a, bv, bacc[mt]);
    }
  }
  // h2f = lrelu(c2b + b_c2b + sc + b_sc) -> stage rows 0..63
#pragma unroll
  for (int mt = 0; mt < 4; ++mt)
#pragma unroll
    for (int r = 0; r < 8; ++r) {
      const int m = mt * 16 + kh * 8 + r;
      stage[wv][m][lm] = lrelu(bacc[mt][r] + c2bb[m] + scacc[mt][r] + scb[m]);
    }

  // --- c3 head (3x64) + residual; lanes 0..15 handle their column ---
  if (kh == 0) {
#pragma unroll
    for (int r = 0; r < 3; ++r) {
      float s = c3b[r];
      for (int m = 0; m < 64; ++m) s += c3w[r * 64 + m] * stage[wv][m][lm];
      pos_out[((size_t)b * 3 + r) * KK + t] =
          posk[((size_t)b * 3 + r) * KK + t] + s;
    }
  }
}

// =====================================================================
// launch
// =====================================================================
extern "C" void kernel_launch(void* const* d_in, const int* in_sizes, int n_in,
                              void* d_out, int out_size, void* d_ws, size_t ws_size,
                              hipStream_t stream) {
  (void)in_sizes; (void)n_in; (void)out_size; (void)ws_size;
  // inputs: pos, f_local, x, then params in _PARAM_SHAPES insertion order, then k
  const float* pos      = (const float*)d_in[0];
  const float* f_local  = (const float*)d_in[1];
  const float* x        = (const float*)d_in[2];
  const float* score_w1 = (const float*)d_in[3];
  const float* score_b1 = (const float*)d_in[4];
  const float* score_w2 = (const float*)d_in[5];
  const float* d1_sc_w  = (const float*)d_in[6];
  const float* d1_sc_b  = (const float*)d_in[7];
  const float* d1_c1_w  = (const float*)d_in[8];
  const float* d1_c1_b  = (const float*)d_in[9];
  const float* d1_c2a_w = (const float*)d_in[10];
  const float* d1_c2a_b = (const float*)d_in[11];
  const float* d1_c2b_w = (const float*)d_in[12];
  const float* d1_c2b_b = (const float*)d_in[13];
  const float* d2_sc_w  = (const float*)d_in[14];
  const float* d2_sc_b  = (const float*)d_in[15];
  const float* d2_c1_w  = (const float*)d_in[16];
  const float* d2_c1_b  = (const float*)d_in[17];
  const float* d2_c2a_w = (const float*)d_in[18];
  const float* d2_c2a_b = (const float*)d_in[19];
  const float* d2_c2b_w = (const float*)d_in[20];
  const float* d2_c2b_b = (const float*)d_in[21];
  const float* c3_w     = (const float*)d_in[22];
  const float* c3_b     = (const float*)d_in[23];

  // outputs, concatenated: pos_out (8,3,8192) | f_local_k (8,128,8192) | feat_k (8,640,8192)
  float* out       = (float*)d_out;
  float* out_pos   = out;
  float* out_floc  = out + (size_t)BB * 3  * KK;
  float* out_feat  = out + (size_t)BB * 3  * KK + (size_t)BB * DL * KK;

  // workspace layout (floats)
  float*    ws     = (float*)d_ws;
  float*    scoreB = ws;                              // B*N
  int*      idxk   = (int*)(ws + (size_t)BB * NN);    // B*K
  float*    sck    = ws + (size_t)BB * NN + (size_t)BB * KK;       // B*K
  float*    posk   = sck + (size_t)BB * KK;                        // B*3*K
  unsigned* gbits  = (unsigned*)(posk + (size_t)BB * 3 * KK);      // B*128
  float*    gvec   = (float*)(gbits + BB * 128);                   // B*128

  // 1) scores (WMMA GEMM fused with w2 dot + sigmoid)
  score_kernel<<<(BB * NN) / 128, 256, 0, stream>>>(x, score_w1, score_b1, score_w2, scoreB);

  // 2) exact ordered top-k: per-batch in-LDS bitonic sort
  topk_sort_kernel<<<BB, 1024, 0, stream>>>(scoreB, idxk, sck);

  // 3) max-pool accumulator init, gather + d1 MLP + pooled max, finalize g
  init_g_kernel<<<(BB * 128 + 255) / 256, 256, 0, stream>>>(gbits);
  gather_d1_kernel<<<(BB * KK) / 256, 256, 0, stream>>>(
      pos, f_local, x, idxk, sck,
      d1_sc_w, d1_sc_b, d1_c1_w, d1_c1_b, d1_c2a_w, d1_c2a_b, d1_c2b_w, d1_c2b_b,
      posk, out_floc, out_feat, gbits);
  fin_g_kernel<<<(BB * 128 + 255) / 256, 256, 0, stream>>>(gbits, gvec);

  // 4) d2 MLP (WMMA) + c3 head + residual
  d2_kernel<<<(BB * KK) / 128, 256, 0, stream>>>(
      posk, gvec, out_feat,
      d2_sc_w, d2_sc_b, d2_c1_w, d2_c1_b, d2_c2a_w, d2_c2a_b, d2_c2b_w, d2_c2b_b,
      c3_w, c3_b, out_pos);
}